// OverlapAdd_14559939133573
// MI455X (gfx1250) — compile-verified
//
#include <hip/hip_runtime.h>
#include <stdint.h>

#define N_FFT   2048
#define HOP     512
#define PAD     1536          // (N_FFT/HOP - 1) * HOP
#define T_IN    1048576
#define BATCH   16
#define FRAMES  2048          // (T_IN + PAD - N_FFT)/HOP + 1
#define THREADS 512           // 512 lanes x float4 = 2048 floats = one frame

// Async framing kernel: one workgroup per (batch, frame).
//   global --(global_load_async_to_lds_b128)--> LDS --(global_store_async_from_lds_b128)--> global
// Data never touches VGPRs; dependencies tracked with ASYNCcnt.
__global__ __launch_bounds__(THREADS)
void frame_gather_async(const float* __restrict__ x, float* __restrict__ out) {
    __shared__ float smem[N_FFT];                 // 8 KB staging per workgroup

    const int bf = blockIdx.x;                    // b * FRAMES + f
    const int b  = bf >> 11;                      // / 2048
    const int f  = bf & (FRAMES - 1);
    const int elem = threadIdx.x * 4;             // element index within frame [0, 2048)

    // Source element index into x[b, :]; negative => zero-pad region.
    const int srcIdx = f * HOP - PAD + elem;      // max = 2047*512-1536+2044 = 1048572 < T_IN

    // 32-bit LDS byte offset of this lane's slot (flat shared addr bits [31:0]).
    const unsigned ldsAddr = (unsigned)(uintptr_t)(&smem[elem]);

    if (srcIdx >= 0) {
        const unsigned long long srcRow =
            (unsigned long long)(x + (size_t)b * T_IN);   // uniform -> SGPR pair
        const unsigned voff = (unsigned)(srcIdx * 4);     // per-lane byte offset
        asm volatile("global_load_async_to_lds_b128 %0, %1, %2"
                     :: "v"(ldsAddr), "v"(voff), "s"(srcRow)
                     : "memory");
    } else {
        // Leading pad: zero this lane's LDS slot directly (aligned float4).
        float4 z = make_float4(0.f, 0.f, 0.f, 0.f);
        *(float4*)(&smem[elem]) = z;
    }

    // RAW through LDS: async loads (ASYNCcnt) and the pad ds_store (DScnt)
    // must land before the async store reads LDS.
    asm volatile("s_wait_asynccnt 0x0\n\t"
                 "s_wait_dscnt 0x0" ::: "memory");

    // Contiguous 8 KB frame in the output; non-temporal (output is write-once,
    // keep the reused 64 MiB input resident in L2 instead).
    const unsigned long long dstRow =
        (unsigned long long)(out + (size_t)bf * N_FFT);   // uniform -> SGPR pair
    const unsigned dvoff = (unsigned)(elem * 4);
    asm volatile("global_store_async_from_lds_b128 %0, %1, %2 th:TH_STORE_NT"
                 :: "v"(dvoff), "v"(ldsAddr), "s"(dstRow)
                 : "memory");
    // s_endpgm performs an implicit wait-idle, so the async store completes
    // before the wave retires and LDS is deallocated.
}

extern "C" void kernel_launch(void* const* d_in, const int* in_sizes, int n_in,
                              void* d_out, int out_size, void* d_ws, size_t ws_size,
                              hipStream_t stream) {
    (void)in_sizes; (void)n_in; (void)out_size; (void)d_ws; (void)ws_size;
    const float* x = (const float*)d_in[0];
    float* out = (float*)d_out;

    dim3 grid(BATCH * FRAMES);   // 32768 workgroups, one frame each
    dim3 block(THREADS);
    frame_gather_async<<<grid, block, 0, stream>>>(x, out);
}